// SingleNode_Reserve_RT_OptNet_46926812676868
// MI455X (gfx1250) — compile-verified
//
#include <hip/hip_runtime.h>

#define BDIM 4096
#define GDIM 100
#define TDIM 24
#define NS   (GDIM + 1)          // 101 sorted offers (incl. slack)
#define NCOL (BDIM * TDIM)       // 98304 columns per side

typedef __attribute__((ext_vector_type(2))) float v2f;
typedef __attribute__((ext_vector_type(8))) float v8f;

// ---------------- Kernel 1: stable price sort (batch-invariant) ----------------
// order[side*NS + r] = original index (0..99 = generator, 100 = slack)
// psort[side*NS + r] = price at sorted position r
__global__ void prep_sort_kernel(const float* __restrict__ bG,
                                 const float* __restrict__ voll,
                                 const float* __restrict__ vosp,
                                 const float* __restrict__ rup,
                                 const float* __restrict__ rdn,
                                 int*   __restrict__ order,
                                 float* __restrict__ psort)
{
    int i = threadIdx.x;
    for (int side = 0; side < 2; ++side) {
        float ratio = side ? rdn[0] : rup[0];
        float slack = side ? vosp[0] : voll[0];
        if (i < NS) {
            float pi = (i < GDIM) ? ratio * bG[i] : slack;
            int rank = 0;
            for (int j = 0; j < NS; ++j) {
                float pj = (j < GDIM) ? ratio * bG[j] : slack;
                rank += (pj < pi) || ((pj == pi) && (j < i));   // stable, matches jnp.argsort
            }
            order[side * NS + rank] = i;
            psort[side * NS + rank] = pi;
        }
    }
}

// ---------------- Kernel 2: greedy merit-order dispatch (bandwidth-bound scan) ----------------
// One thread per (side, b, t) column. Prefix state stays in registers.
__global__ void dispatch_kernel(const float* __restrict__ R_up,
                                const float* __restrict__ R_dn,
                                const float* __restrict__ omega,
                                const int*   __restrict__ order,
                                const float* __restrict__ psort,
                                float* __restrict__ du, float* __restrict__ dd,
                                float* __restrict__ LS, float* __restrict__ SP,
                                float* __restrict__ obj)   // [2*NCOL] partials (may be null)
{
    __shared__ int   s_order[2 * NS];
    __shared__ float s_price[2 * NS];
    for (int k = threadIdx.x; k < 2 * NS; k += blockDim.x) {
        s_order[k] = order[k];
        s_price[k] = psort[k];
    }
    __syncthreads();

    int col  = blockIdx.x * blockDim.x + threadIdx.x;  // 0 .. 2*NCOL-1 (exact)
    int side = (col >= NCOL) ? 1 : 0;
    int bt   = side ? (col - NCOL) : col;
    int b    = bt / TDIM;
    int t    = bt - b * TDIM;

    float om = omega[bt];
    float d  = side ? fmaxf(-om, 0.0f) : fmaxf(om, 0.0f);

    const float* caps = side ? R_dn : R_up;
    float*       out  = side ? dd   : du;
    const int*   so   = s_order + side * NS;
    const float* sp   = s_price + side * NS;

    size_t base = (size_t)b * (GDIM * TDIM) + (size_t)t;
    float run = 0.0f, ob = 0.0f;

    #pragma unroll 1
    for (int r = 0; r < NS; ++r) {
        int   g = so[r];
        float p = sp[r];
        if (g < GDIM) {
            size_t idx = base + (size_t)g * TDIM;
            float c = caps[idx];
            float a = fminf(fmaxf(d - run, 0.0f), c);  // clip(d - before, 0, cap)
            out[idx] = a;
            ob  = fmaf(p, a, ob);
            run += c;
        } else {                                       // slack offer: cap == demand
            float a = fminf(fmaxf(d - run, 0.0f), d);
            if (side) SP[bt] = a; else LS[bt] = a;
            ob  = fmaf(p, a, ob);                      // p == voll / vosp here
            run += d;
        }
    }
    if (obj) obj[side * NCOL + bt] = ob;
}

// ---------------- Kernel 3: rt_obj via V_WMMA_F32_16X16X4_F32 ----------------
// rt_obj[b] = sum over 48 partials (24 up + 24 dn). Per wave: 16 batches,
// A = 16x4 tile of partials (lanes 0-15 hold K=0,1; lanes 16-31 hold K=2,3),
// B = all-ones 4x16 -> D rows are the running row-sums (all 16 columns equal).
// Column N=0 lives in lane 0 (M=0..7) and lane 16 (M=8..15).
__global__ void obj_reduce_wmma_kernel(const float* __restrict__ obj,
                                       float* __restrict__ rt)
{
    int lane = threadIdx.x;        // 0..31, all active (EXEC must be all ones)
    int m    = lane & 15;          // A-matrix row = batch within tile
    int half = lane >> 4;          // 0: K={kb,kb+1}; 1: K={kb+2,kb+3}
    int bb   = blockIdx.x * 16;

    const float* rowu = obj + (size_t)(bb + m) * TDIM;          // up partials
    const float* rowd = obj + NCOL + (size_t)(bb + m) * TDIM;   // dn partials

    v8f acc = {};
    v2f bones = {1.0f, 1.0f};      // all-ones B matrix (layout-immune)

    #pragma unroll
    for (int kb = 0; kb < 2 * TDIM; kb += 4) {
        int k0 = kb + half * 2;
        int k1 = k0 + 1;
        v2f a;
        a.x = (k0 < TDIM) ? rowu[k0] : rowd[k0 - TDIM];
        a.y = (k1 < TDIM) ? rowu[k1] : rowd[k1 - TDIM];
        acc = __builtin_amdgcn_wmma_f32_16x16x4_f32(
                  /*neg_a=*/false, a, /*neg_b=*/false, bones,
                  /*c_mod=*/(short)0, acc, /*reuse_a=*/false, /*reuse_b=*/false);
    }

    if (m == 0) {                  // lanes 0 and 16 hold column N=0
        int mb = half * 8;
        #pragma unroll
        for (int j = 0; j < 8; ++j) rt[bb + mb + j] = acc[j];
    }
}

// ---------------- Fallback reduction (only if ws too small; deterministic) ----------------
__global__ void obj_reduce_fallback_kernel(const float* __restrict__ du,
                                           const float* __restrict__ dd,
                                           const float* __restrict__ LS,
                                           const float* __restrict__ SP,
                                           const float* __restrict__ bG,
                                           const float* __restrict__ voll,
                                           const float* __restrict__ vosp,
                                           const float* __restrict__ rup,
                                           const float* __restrict__ rdn,
                                           float* __restrict__ rt)
{
    __shared__ float s[256];
    int b = blockIdx.x;
    float cu = rup[0], cd = rdn[0];
    const float* pu = du + (size_t)b * GDIM * TDIM;
    const float* pd = dd + (size_t)b * GDIM * TDIM;
    float acc = 0.0f;
    for (int i = threadIdx.x; i < GDIM * TDIM; i += blockDim.x) {
        int g = i / TDIM;
        float cg = bG[g];
        acc += cu * cg * pu[i] + cd * cg * pd[i];
    }
    for (int t = threadIdx.x; t < TDIM; t += blockDim.x)
        acc += voll[0] * LS[b * TDIM + t] + vosp[0] * SP[b * TDIM + t];
    s[threadIdx.x] = acc;
    __syncthreads();
    for (int off = 128; off > 0; off >>= 1) {
        if (threadIdx.x < off) s[threadIdx.x] += s[threadIdx.x + off];
        __syncthreads();
    }
    if (threadIdx.x == 0) rt[b] = s[0];
}

extern "C" void kernel_launch(void* const* d_in, const int* in_sizes, int n_in,
                              void* d_out, int out_size, void* d_ws, size_t ws_size,
                              hipStream_t stream)
{
    const float* R_up  = (const float*)d_in[0];
    const float* R_dn  = (const float*)d_in[1];
    const float* omega = (const float*)d_in[2];
    const float* bG    = (const float*)d_in[3];
    const float* voll  = (const float*)d_in[4];
    const float* vosp  = (const float*)d_in[5];
    const float* rup   = (const float*)d_in[6];
    const float* rdn   = (const float*)d_in[7];

    float* du = (float*)d_out;
    float* dd = du + (size_t)BDIM * GDIM * TDIM;
    float* LS = dd + (size_t)BDIM * GDIM * TDIM;
    float* SP = LS + (size_t)BDIM * TDIM;
    float* rt = SP + (size_t)BDIM * TDIM;

    // Workspace layout: order[2*NS] (int), psort[2*NS] (float), obj[2*NCOL] (float)
    int*   order = (int*)d_ws;
    float* psort = (float*)(order + 2 * NS);
    float* obj   = psort + 2 * NS;                       // byte offset 1616 (16B aligned)
    size_t need_obj = (size_t)2 * NS * (sizeof(int) + sizeof(float))
                    + (size_t)2 * NCOL * sizeof(float);
    bool have_obj = (ws_size >= need_obj);               // fixed per run -> deterministic path

    prep_sort_kernel<<<1, 128, 0, stream>>>(bG, voll, vosp, rup, rdn, order, psort);

    dispatch_kernel<<<(2 * NCOL) / 256, 256, 0, stream>>>(
        R_up, R_dn, omega, order, psort, du, dd, LS, SP,
        have_obj ? obj : (float*)nullptr);

    if (have_obj) {
        obj_reduce_wmma_kernel<<<BDIM / 16, 32, 0, stream>>>(obj, rt);
    } else {
        obj_reduce_fallback_kernel<<<BDIM, 256, 0, stream>>>(
            du, dd, LS, SP, bG, voll, vosp, rup, rdn, rt);
    }
}